// ODLLoss_37615323578930
// MI455X (gfx1250) — compile-verified
//
#include <hip/hip_runtime.h>
#include <math.h>

#define B_N 8192
#define C_N 100
#define D_N 128
#define NPAD 112              // 7 tiles of 16 (covers C=100)
#define ROWS_PER_BLOCK 64
#define NBLOCKS (B_N / ROWS_PER_BLOCK)   // 128
#define MARGIN_F 20.0f
#define EPS_F 1e-8f

// workspace layout (in floats)
#define WS_S1   0                         // stage-1 scalar term
#define WS_CC   16                        // 112 floats: ||c_j||^2
#define WS_H    128                       // 100*100 floats: centers gram
#define WS_PART (128 + C_N * C_N)         // 128 blocks * 2 floats

typedef __attribute__((ext_vector_type(2))) float v2f;
typedef __attribute__((ext_vector_type(8))) float v8f;

__device__ __forceinline__ float clip_f(float v, float lo, float hi) {
  return fminf(fmaxf(v, lo), hi);
}

// ---------------------------------------------------------------------------
// Stage 1: all center-center quantities (one workgroup; ~1.3 MFLOP total).
// Produces: ws[WS_CC + j] = ||c_j||^2, ws[WS_H + i*100 + j] = c_i . c_j,
//           ws[WS_S1] = 0.1 * (loss_repel + inter_cos)
// ---------------------------------------------------------------------------
__global__ void __launch_bounds__(128) stage1_centers(const float* __restrict__ centers,
                                                      float* __restrict__ ws) {
  __shared__ float Hs[C_N * C_N];   // 40 KB
  __shared__ float ccS[C_N];
  __shared__ float red[128];
  const int tid = threadIdx.x;

  for (int j = tid; j < C_N; j += 128) {
    const float* c = centers + j * D_N;
    float s = 0.f;
    #pragma unroll 8
    for (int d = 0; d < D_N; ++d) s = fmaf(c[d], c[d], s);
    ccS[j] = s;
    ws[WS_CC + j] = s;
  }
  for (int p = tid; p < C_N * C_N; p += 128) {
    const int i = p / C_N, j = p % C_N;
    const float* ci = centers + i * D_N;
    const float* cj = centers + j * D_N;
    float s = 0.f;
    #pragma unroll 8
    for (int d = 0; d < D_N; ++d) s = fmaf(ci[d], cj[d], s);
    Hs[p] = s;
    ws[WS_H + p] = s;
  }
  __syncthreads();

  float contrib = 0.f;
  if (tid < C_N) {
    const int i = tid;
    // nearest other center:  dc[i][j] = cc_i + cc_j - 2 H[i][j]
    float best = 3.4e38f; int n = 0;
    for (int j = 0; j < C_N; ++j) {
      if (j == i) continue;
      const float dcij = ccS[i] + ccS[j] - 2.f * Hs[i * C_N + j];
      if (dcij < best) { best = dcij; n = j; }
    }
    const float repel = clip_f(MARGIN_F - best, 1e-12f, 1e12f);
    float n1 = sqrtf(best > 0.f ? best : 1.f);
    n1 = fmaxf(n1, EPS_F);
    // stable softmax over -dis_nn, mask j != i && j != n
    float m = 3.4e38f;
    for (int j = 0; j < C_N; ++j) {
      if (j == i || j == n) continue;
      const float dnn = ccS[n] + ccS[j] - 2.f * Hs[n * C_N + j];
      m = fminf(m, dnn);
    }
    float S = 0.f, T = 0.f;
    for (int j = 0; j < C_N; ++j) {
      if (j == i || j == n) continue;
      const float dnn = ccS[n] + ccS[j] - 2.f * Hs[n * C_N + j];
      const float w = expf(m - dnn);
      const float dcij = ccS[i] + ccS[j] - 2.f * Hs[i * C_N + j];
      float n2 = sqrtf(dcij > 0.f ? dcij : 1.f);
      n2 = fmaxf(n2, EPS_F);
      // (c_i - c_n) . (c_i - c_j) via gram identity
      const float num = ccS[i] - Hs[i * C_N + j] - Hs[n * C_N + i] + Hs[n * C_N + j];
      const float cosn = num / (n1 * n2);
      S += w;
      T = fmaf(1.f + cosn, w, T);
    }
    contrib = repel + T / S;
  }
  red[tid] = contrib;
  __syncthreads();
  for (int s = 64; s > 0; s >>= 1) {
    if (tid < s) red[tid] += red[tid + s];
    __syncthreads();
  }
  if (tid == 0) ws[WS_S1] = 0.1f * red[0] / (float)C_N;
}

// ---------------------------------------------------------------------------
// Stage 2: G = x @ centers^T via V_WMMA_F32_16X16X4_F32, then per-row epilogue.
// Block = 128 threads = 4 waves; each wave owns a 16-row M tile.
// x is read from HBM exactly once: the WMMA A-operands also feed ||x||^2.
// ---------------------------------------------------------------------------
__global__ void __launch_bounds__(128) stage2_samples(const float* __restrict__ x,
                                                      const int* __restrict__ labels,
                                                      const float* __restrict__ centers,
                                                      float* __restrict__ ws) {
  __shared__ float Gt[ROWS_PER_BLOCK * NPAD];   // 28 KB
  __shared__ float xxP0[ROWS_PER_BLOCK];        // ||x||^2 partial, lanes 0-15
  __shared__ float xxP1[ROWS_PER_BLOCK];        // ||x||^2 partial, lanes 16-31
  __shared__ float redC[128];
  __shared__ float redD[128];

  const int tid  = threadIdx.x;
  const int lane = tid & 31;
  const int wave = tid >> 5;
  const int half = lane >> 4;   // 0: lanes 0-15 (K = 4k..4k+1), 1: lanes 16-31 (K = 4k+2..4k+3)
  const int l16  = lane & 15;

  // ---- preload A operands: 16x4 f32 tiles, 32 K-steps -> 32 x float2/lane ----
  const int rowA = blockIdx.x * ROWS_PER_BLOCK + wave * 16 + l16;
  const v2f* __restrict__ xa = (const v2f*)(x + (size_t)rowA * D_N);
  v2f a[32];
  #pragma unroll
  for (int k = 0; k < 32; ++k) a[k] = xa[2 * k + half];

  // ||x_row||^2 from the already-resident A registers (no second HBM pass):
  // lane L holds 64 of the row's 128 elements, lane L+16 holds the rest.
  {
    float ss = 0.f;
    #pragma unroll
    for (int k = 0; k < 32; ++k) {
      ss = fmaf(a[k].x, a[k].x, ss);
      ss = fmaf(a[k].y, a[k].y, ss);
    }
    if (half == 0) xxP0[wave * 16 + l16] = ss;
    else           xxP1[wave * 16 + l16] = ss;
  }

  // ---- 7 N-tiles of 16 centers each. Columns >= 100 clamp to row 0 (finite
  // garbage lands in padded Gt columns that the epilogue never reads). ----
  for (int nt = 0; nt < 7; ++nt) {
    const int nCol = nt * 16 + l16;
    const int nColC = (nCol < C_N) ? nCol : 0;
    const v2f* __restrict__ cb = (const v2f*)(centers + (size_t)nColC * D_N);
    v8f acc = {};
    #pragma unroll
    for (int k = 0; k < 32; ++k) {
      const v2f bv = cb[2 * k + half];
      acc = __builtin_amdgcn_wmma_f32_16x16x4_f32(false, a[k], false, bv,
                                                  (short)0, acc, false, false);
    }
    // C/D layout: VGPR r holds rows r (lanes 0-15) and r+8 (lanes 16-31), N = lane%16
    #pragma unroll
    for (int r = 0; r < 8; ++r) {
      Gt[(wave * 16 + r + 8 * half) * NPAD + nt * 16 + l16] = acc[r];
    }
  }
  __syncthreads();

  // ---- per-row loss terms: threads 0..63 each own one sample row ----
  float lcen = 0.f, ldir = 0.f;
  if (tid < ROWS_PER_BLOCK) {
    const int brow = blockIdx.x * ROWS_PER_BLOCK + tid;
    const int l = labels[brow];
    const float xx = xxP0[tid] + xxP1[tid];

    const float* Grow = &Gt[tid * NPAD];
    const float* cc = ws + WS_CC;
    const float* Hl = ws + WS_H + l * C_N;

    const float Gl = Grow[l];
    const float distl = xx + cc[l] - 2.f * Gl;
    // reference clips the full onehot-masked row -> zeros clip to 1e-12 each
    lcen = clip_f(distl, 1e-12f, 1e12f) + (float)(C_N - 1) * 1e-12f;
    float nl = sqrtf(distl > 0.f ? distl : 1.f);
    nl = fmaxf(nl, EPS_F);

    float m = 3.4e38f;
    for (int j = 0; j < C_N; ++j) {
      if (j == l) continue;
      const float dj = xx + cc[j] - 2.f * Grow[j];
      m = fminf(m, dj);
    }
    float S = 0.f, T = 0.f;
    for (int j = 0; j < C_N; ++j) {
      if (j == l) continue;
      const float Gj = Grow[j];
      const float dj = xx + cc[j] - 2.f * Gj;
      const float w = expf(m - dj);
      float nj = sqrtf(dj > 0.f ? dj : 1.f);
      nj = fmaxf(nj, EPS_F);
      // (c_l - x).(c_j - x) = H[l][j] - G[b][j] - G[b][l] + ||x||^2
      const float num = Hl[j] - Gj - Gl + xx;
      const float cd = num / (nl * nj);
      S += w;
      T = fmaf(1.f - cd, w, T);
    }
    ldir = T / S;
  }
  redC[tid] = lcen;
  redD[tid] = ldir;
  __syncthreads();
  for (int s = 64; s > 0; s >>= 1) {
    if (tid < s) { redC[tid] += redC[tid + s]; redD[tid] += redD[tid + s]; }
    __syncthreads();
  }
  if (tid == 0) {
    ws[WS_PART + 2 * blockIdx.x + 0] = redC[0];
    ws[WS_PART + 2 * blockIdx.x + 1] = redD[0];
  }
}

// ---------------------------------------------------------------------------
// Stage 3: deterministic final reduction over the 128 block partials.
// ---------------------------------------------------------------------------
__global__ void __launch_bounds__(128) stage3_reduce(const float* __restrict__ ws,
                                                     float* __restrict__ out) {
  __shared__ float rC[128];
  __shared__ float rD[128];
  const int tid = threadIdx.x;
  rC[tid] = ws[WS_PART + 2 * tid + 0];
  rD[tid] = ws[WS_PART + 2 * tid + 1];
  __syncthreads();
  for (int s = 64; s > 0; s >>= 1) {
    if (tid < s) { rC[tid] += rC[tid + s]; rD[tid] += rD[tid + s]; }
    __syncthreads();
  }
  if (tid == 0) {
    out[0] = rC[0] / (float)B_N + rD[0] / (float)B_N + ws[WS_S1];
  }
}

extern "C" void kernel_launch(void* const* d_in, const int* in_sizes, int n_in,
                              void* d_out, int out_size, void* d_ws, size_t ws_size,
                              hipStream_t stream) {
  (void)in_sizes; (void)n_in; (void)out_size; (void)ws_size;
  const float* x       = (const float*)d_in[0];
  const int*   labels  = (const int*)d_in[1];
  const float* centers = (const float*)d_in[2];
  float* ws  = (float*)d_ws;
  float* out = (float*)d_out;

  stage1_centers<<<1, 128, 0, stream>>>(centers, ws);
  stage2_samples<<<NBLOCKS, 128, 0, stream>>>(x, labels, centers, ws);
  stage3_reduce<<<1, 128, 0, stream>>>(ws, out);
}